// GCN_20495583937244
// MI455X (gfx1250) — compile-verified
//
#include <hip/hip_runtime.h>
#include <hip/hip_bf16.h>

typedef __attribute__((ext_vector_type(2))) float v2f;
typedef __attribute__((ext_vector_type(8))) float v8f;

#define NN 50000
#define NE 800000
#define HIDC 128
#define OUTC 40
#define EPSF 1e-5f

// ---------------- utility kernels ----------------

__global__ void k_zero(float* __restrict__ p, int n) {
    int i = blockIdx.x * blockDim.x + threadIdx.x;
    if (i < n) p[i] = 0.0f;
}

__global__ void k_degree(const int* __restrict__ dst, float* __restrict__ deg, int e) {
    int i = blockIdx.x * blockDim.x + threadIdx.x;
    if (i < e) atomicAdd(&deg[dst[i]], 1.0f);
}

__global__ void k_dinv(float* __restrict__ deg, int n) {
    int i = blockIdx.x * blockDim.x + threadIdx.x;
    if (i < n) {
        float d = deg[i];
        deg[i] = (d > 0.0f) ? rsqrtf(fmaxf(d, 1.0f)) : 0.0f;
    }
}

__global__ void k_norm(const int* __restrict__ ei, const float* __restrict__ dinv,
                       float* __restrict__ nrm, int e) {
    int i = blockIdx.x * blockDim.x + threadIdx.x;
    if (i < e) nrm[i] = dinv[ei[i]] * dinv[ei[NE + i]];
}

// ---------------- WMMA GEMM: C[n x 128] = A[n x 128] * W[128 x 128] ----------------
// block = 256 threads = 8 wave32; each wave computes a 16-row strip x 128 cols.
// V_WMMA_F32_16X16X4_F32 layouts (ISA 7.12.2):
//  A 16x4 : lane m=lane&15, kh=lane>>4; vgpr v holds A[m][kh*2+v]
//  B 4x16 : lane n=lane&15, kh=lane>>4; vgpr v holds B[kh*2+v][n]
//  C 16x16: vgpr r holds C[r + 8*kh][n]
// W is staged in LDS with row-pairs interleaved: wl[(p*128+n)*2 + j] = W[2p+j][n],
// so a B fragment is one 8B-aligned ds_load_b64 straight into an even VGPR pair
// (no repack moves). A-row m only feeds C-row m and out-of-range C rows are
// dropped at the store, so the A load uses a clamped, always in-bounds row.
__global__ __launch_bounds__(256) void k_gemm128(const float* __restrict__ A,
                                                 const float* __restrict__ W,
                                                 float* __restrict__ C, int nrows) {
    __shared__ float wl[HIDC * HIDC];
    for (int i = threadIdx.x; i < 64 * HIDC; i += 256) {
        int p = i >> 7;          // row pair
        int n = i & 127;         // column
        v2f w;
        w.x = W[(2 * p) * HIDC + n];
        w.y = W[(2 * p + 1) * HIDC + n];
        *(v2f*)&wl[(size_t)i * 2] = w;
    }
    __syncthreads();

    const int wave = threadIdx.x >> 5;
    const int lane = threadIdx.x & 31;
    const int m    = lane & 15;
    const int kh   = lane >> 4;          // 0 or 1
    const int rowBase = blockIdx.x * 128 + wave * 16;
    int arow = rowBase + m;
    if (arow >= nrows) arow = nrows - 1;           // clamp: load stays in-bounds
    const float* __restrict__ arowp = A + (size_t)arow * HIDC + kh * 2;

    v8f acc[8] = {};

    for (int k = 0; k < HIDC; k += 4) {
        v2f a = *(const v2f*)(arowp + k);          // 8B-aligned, unconditional
        // pair-row index for this fragment: k/2 + kh
        const float* __restrict__ b0 = &wl[((k >> 1) + kh) * (HIDC * 2) + m * 2];
#pragma unroll
        for (int t = 0; t < 8; ++t) {
            v2f b = *(const v2f*)(b0 + t * 32);    // single ds_load_b64
            acc[t] = __builtin_amdgcn_wmma_f32_16x16x4_f32(
                false, a, false, b, (short)0, acc[t], false, false);
        }
    }

#pragma unroll
    for (int t = 0; t < 8; ++t) {
#pragma unroll
        for (int r = 0; r < 8; ++r) {
            int row = rowBase + r + 8 * kh;
            if (row < nrows) C[(size_t)row * HIDC + t * 16 + m] = acc[t][r];
        }
    }
}

// ---------------- WMMA GEMM: C[n x 40] = A[n x 128] * W[128 x 40] ----------------
__global__ __launch_bounds__(256) void k_gemm40(const float* __restrict__ A,
                                                const float* __restrict__ W,
                                                float* __restrict__ C, int nrows) {
    __shared__ float wl[HIDC * OUTC];    // pair-interleaved: wl[(p*40+n)*2+j] = W[2p+j][n]
    for (int i = threadIdx.x; i < 64 * OUTC; i += 256) {
        int p = i / OUTC;
        int n = i - p * OUTC;
        v2f w;
        w.x = W[(2 * p) * OUTC + n];
        w.y = W[(2 * p + 1) * OUTC + n];
        *(v2f*)&wl[(size_t)i * 2] = w;
    }
    __syncthreads();

    const int wave = threadIdx.x >> 5;
    const int lane = threadIdx.x & 31;
    const int m    = lane & 15;
    const int kh   = lane >> 4;
    const int rowBase = blockIdx.x * 128 + wave * 16;
    int arow = rowBase + m;
    if (arow >= nrows) arow = nrows - 1;
    const float* __restrict__ arowp = A + (size_t)arow * HIDC + kh * 2;

    v8f acc[3] = {};

    for (int k = 0; k < HIDC; k += 4) {
        v2f a = *(const v2f*)(arowp + k);
        const float* __restrict__ b0 = &wl[((k >> 1) + kh) * (OUTC * 2)];
#pragma unroll
        for (int t = 0; t < 3; ++t) {
            int col = t * 16 + m;
            v2f b = {0.0f, 0.0f};
            if (col < OUTC) b = *(const v2f*)(b0 + col * 2);   // ds_load_b64
            acc[t] = __builtin_amdgcn_wmma_f32_16x16x4_f32(
                false, a, false, b, (short)0, acc[t], false, false);
        }
    }

#pragma unroll
    for (int t = 0; t < 3; ++t) {
        int col = t * 16 + m;
        if (col < OUTC) {
#pragma unroll
            for (int r = 0; r < 8; ++r) {
                int row = rowBase + r + 8 * kh;
                if (row < nrows) C[(size_t)row * OUTC + col] = acc[t][r];
            }
        }
    }
}

// ---------------- edge scatter: O[dst] += norm * T[src]  (128 feats) ----------------
// one thread per (edge, 4 features): tid>>5 = edge, (tid&31)*4 = feature chunk
__global__ void k_scatter128(const int* __restrict__ ei, const float* __restrict__ nrm,
                             const float* __restrict__ T, float* __restrict__ O, int total) {
    int tid = blockIdx.x * blockDim.x + threadIdx.x;
    if (tid >= total) return;
    int e  = tid >> 5;
    int c4 = (tid & 31) << 2;
    int s = ei[e];
    int d = ei[NE + e];
    float w = nrm[e];
    float4 v = *(const float4*)(T + (size_t)s * HIDC + c4);
    float* o = O + (size_t)d * HIDC + c4;
    atomicAdd(o + 0, w * v.x);
    atomicAdd(o + 1, w * v.y);
    atomicAdd(o + 2, w * v.z);
    atomicAdd(o + 3, w * v.w);
}

// one thread per (edge, 4 features) for 40-wide output
__global__ void k_scatter40(const int* __restrict__ ei, const float* __restrict__ nrm,
                            const float* __restrict__ T, float* __restrict__ O, int total) {
    int tid = blockIdx.x * blockDim.x + threadIdx.x;
    if (tid >= total) return;
    int e  = tid / 10;
    int c4 = (tid - e * 10) << 2;
    int s = ei[e];
    int d = ei[NE + e];
    float w = nrm[e];
    float4 v = *(const float4*)(T + (size_t)s * OUTC + c4);
    float* o = O + (size_t)d * OUTC + c4;
    atomicAdd(o + 0, w * v.x);
    atomicAdd(o + 1, w * v.y);
    atomicAdd(o + 2, w * v.z);
    atomicAdd(o + 3, w * v.w);
}

// ---------------- BatchNorm ----------------

__global__ __launch_bounds__(128) void k_bnstats(const float* __restrict__ A,
                                                 float* __restrict__ gsum,
                                                 float* __restrict__ gsq, int nrows) {
    int c  = threadIdx.x;              // channel
    int r0 = blockIdx.x * 64;
    int r1 = min(r0 + 64, nrows);
    float s = 0.0f, q = 0.0f;
    for (int r = r0; r < r1; ++r) {
        float v = A[(size_t)r * HIDC + c];
        s += v;
        q += v * v;
    }
    atomicAdd(&gsum[c], s);
    atomicAdd(&gsq[c], q);
}

__global__ void k_bnfinal(const float* __restrict__ gsum, const float* __restrict__ gsq,
                          const float* __restrict__ g, const float* __restrict__ beta,
                          float* __restrict__ scale, float* __restrict__ shift) {
    int c = threadIdx.x;               // 128 threads
    float inv = 1.0f / (float)NN;
    float mu  = gsum[c] * inv;
    float var = gsq[c] * inv - mu * mu;
    float sc  = g[c] * rsqrtf(var + EPSF);
    scale[c] = sc;
    shift[c] = beta[c] - mu * sc;
}

__global__ void k_bnapply(const float* __restrict__ A, const float* __restrict__ scale,
                          const float* __restrict__ shift, float* __restrict__ H, int total) {
    int i = blockIdx.x * blockDim.x + threadIdx.x;
    if (i < total) {
        int c = i & (HIDC - 1);
        H[i] = fmaxf(fmaf(A[i], scale[c], shift[c]), 0.0f);
    }
}

__global__ void k_initout(float* __restrict__ out, const float* __restrict__ b3, int total) {
    int i = blockIdx.x * blockDim.x + threadIdx.x;
    if (i < total) out[i] = b3[i % OUTC];
}

// ---------------- host orchestration ----------------

extern "C" void kernel_launch(void* const* d_in, const int* in_sizes, int n_in,
                              void* d_out, int out_size, void* d_ws, size_t ws_size,
                              hipStream_t stream) {
    const float* x    = (const float*)d_in[0];
    const int*   ei   = (const int*)d_in[1];     // [2,E] row-major: src row 0, dst row 1
    const float* W1   = (const float*)d_in[2];
    const float* g1   = (const float*)d_in[4];
    const float* be1  = (const float*)d_in[5];
    const float* W2   = (const float*)d_in[6];
    const float* g2   = (const float*)d_in[8];
    const float* be2  = (const float*)d_in[9];
    const float* W3   = (const float*)d_in[10];
    const float* b3   = (const float*)d_in[11];
    float* out = (float*)d_out;

    // workspace layout (floats)
    float* ws    = (float*)d_ws;
    float* nrm   = ws;                           // E
    float* deg   = nrm + NE;                     // N (becomes dinv in place)
    float* gsum  = deg + NN;                     // 128
    float* gsq   = gsum + HIDC;                  // 128
    float* scale = gsq + HIDC;                   // 128
    float* shift = scale + HIDC;                 // 128
    float* bufT  = shift + HIDC;                 // N*128
    float* bufA  = bufT + (size_t)NN * HIDC;     // N*128

    const int B = 256;
    const int gN      = (NN + B - 1) / B;
    const int gE      = (NE + B - 1) / B;
    const int gElem   = (NN * HIDC + B - 1) / B;
    const int gGemm   = (NN + 127) / 128;
    const int sc128   = NE * 32;
    const int gSc128  = (sc128 + B - 1) / B;
    const int sc40    = NE * 10;
    const int gSc40   = (sc40 + B - 1) / B;
    const int gStats  = (NN + 63) / 64;
    const int gOut    = (NN * OUTC + B - 1) / B;

    // ---- gcn_norm (shared across layers) ----
    k_zero<<<gN, B, 0, stream>>>(deg, NN);
    k_degree<<<gE, B, 0, stream>>>(ei + NE, deg, NE);
    k_dinv<<<gN, B, 0, stream>>>(deg, NN);
    k_norm<<<gE, B, 0, stream>>>(ei, deg, nrm, NE);

    // ---- layer 1: T1 = X@W1 ; A1 = scatter(T1) ; H1 = BNReLU(A1) -> bufT ----
    k_gemm128<<<gGemm, B, 0, stream>>>(x, W1, bufT, NN);
    k_zero<<<gElem, B, 0, stream>>>(bufA, NN * HIDC);
    k_scatter128<<<gSc128, B, 0, stream>>>(ei, nrm, bufT, bufA, sc128);
    k_zero<<<1, B, 0, stream>>>(gsum, 2 * HIDC);
    k_bnstats<<<gStats, 128, 0, stream>>>(bufA, gsum, gsq, NN);
    k_bnfinal<<<1, 128, 0, stream>>>(gsum, gsq, g1, be1, scale, shift);
    k_bnapply<<<gElem, B, 0, stream>>>(bufA, scale, shift, bufT, NN * HIDC);

    // ---- layer 2: T2 = H1@W2 -> bufA ; A2 = scatter(T2) -> bufT ; H2 -> bufA ----
    k_gemm128<<<gGemm, B, 0, stream>>>(bufT, W2, bufA, NN);
    k_zero<<<gElem, B, 0, stream>>>(bufT, NN * HIDC);
    k_scatter128<<<gSc128, B, 0, stream>>>(ei, nrm, bufA, bufT, sc128);
    k_zero<<<1, B, 0, stream>>>(gsum, 2 * HIDC);
    k_bnstats<<<gStats, 128, 0, stream>>>(bufT, gsum, gsq, NN);
    k_bnfinal<<<1, 128, 0, stream>>>(gsum, gsq, g2, be2, scale, shift);
    k_bnapply<<<gElem, B, 0, stream>>>(bufT, scale, shift, bufA, NN * HIDC);

    // ---- layer 3: T3 = H2@W3 -> bufT(40-wide) ; out = b3 + scatter(T3) ----
    k_gemm40<<<gGemm, B, 0, stream>>>(bufA, W3, bufT, NN);
    k_initout<<<gOut, B, 0, stream>>>(out, b3, NN * OUTC);
    k_scatter40<<<gSc40, B, 0, stream>>>(ei, nrm, bufT, out, sc40);
}